// ABP_13159779795098
// MI455X (gfx1250) — compile-verified
//
#include <hip/hip_runtime.h>
#include <stdint.h>

// Problem constants (fixed by the reference):
#define NS   8
#define B_   16
#define C_   512
#define H_   128
#define W_   128
#define HW_  (H_ * W_)        // 16384 elements = 64 KB per (b,c) slab

// ---------------------------------------------------------------------------
// Kernel 0: zero the per-(b,h) argmax-row-count accumulator (workspace is
// poisoned between runs, so this must run every call).
// ---------------------------------------------------------------------------
__global__ void __launch_bounds__(256) zero_rc_kernel(int* __restrict__ rc, int n) {
    int i = blockIdx.x * 256 + threadIdx.x;
    if (i < n) rc[i] = 0;
}

// ---------------------------------------------------------------------------
// Kernel 1: one block per (b,c). Stage the 64 KB slab into LDS via the CDNA5
// async global->LDS path (one HBM read of the 512 MB input total), then:
//   pass 1: block-wide max (vmax per (b,c))
//   pass 2: slab sum (-> F[b,c] = sum/W) + per-row tie counts vs vmax,
//           accumulated into rc[b,h] with integer atomics (deterministic).
// ---------------------------------------------------------------------------
__global__ void __launch_bounds__(256) stats_kernel(const float* __restrict__ x,
                                                    float* __restrict__ Fout,   // [B*C]
                                                    int*   __restrict__ rc) {   // [B*H]
    extern __shared__ char smraw[];
    float* sm  = (float*)smraw;                                   // 16384 floats (64 KB)
    int*   cnt = (int*)  (smraw + HW_ * sizeof(float));           // 128 ints
    float* red = (float*)(smraw + HW_ * sizeof(float) + H_ * sizeof(int)); // 256 floats

    const int tid = threadIdx.x;          // 0..255
    const int bc  = blockIdx.x;           // b*C + c
    const float* src = x + (size_t)bc * HW_;

    // ---- async copy: 16 issues x (256 lanes x 16 B) = 64 KB, fully coalesced
    const unsigned lds0 = (unsigned)(uintptr_t)smraw;   // low 32 bits = LDS byte offset
    const uint64_t g0   = (uint64_t)(uintptr_t)src;
#pragma unroll
    for (int i = 0; i < 16; ++i) {
        unsigned off = (unsigned)(i * 4096 + tid * 16);  // bytes
        unsigned lAddr = lds0 + off;
        uint64_t gAddr = g0 + (uint64_t)off;
        asm volatile("global_load_async_to_lds_b128 %0, %1, off"
                     :: "v"(lAddr), "v"(gAddr)
                     : "memory");
    }
    if (tid < H_) cnt[tid] = 0;
    asm volatile("s_wait_asynccnt 0x0" ::: "memory");
    __syncthreads();

    // ---- pass 1: block max. sm[tid + 256k] -> bank = tid%64: conflict-free.
    float m = -__builtin_inff();
#pragma unroll 8
    for (int k = 0; k < 64; ++k) m = fmaxf(m, sm[tid + 256 * k]);
    red[tid] = m;
    __syncthreads();
    for (int s = 128; s > 0; s >>= 1) {
        if (tid < s) red[tid] = fmaxf(red[tid], red[tid + s]);
        __syncthreads();
    }
    const float vmax = red[0];
    __syncthreads();

    // ---- pass 2: sum + tie counts per row (row h = idx>>7). Ties are rare,
    // so the ds_add is almost never executed.
    float fs = 0.0f;
#pragma unroll 8
    for (int k = 0; k < 64; ++k) {
        int idx = tid + 256 * k;
        float v = sm[idx];
        fs += v;
        if (v == vmax) atomicAdd(&cnt[idx >> 7], 1);
    }
    red[tid] = fs;
    __syncthreads();
    for (int s = 128; s > 0; s >>= 1) {
        if (tid < s) red[tid] += red[tid + s];
        __syncthreads();
    }
    if (tid == 0) Fout[bc] = red[0] * (1.0f / (float)W_);   // F[b,c] = sum/W
    if (tid < H_) {
        int cv = cnt[tid];
        if (cv > 0) atomicAdd(&rc[(bc >> 9) * H_ + tid], cv);  // b = bc/C
    }
}

// ---------------------------------------------------------------------------
// Kernel 2: one block per b. Exclusive cumsum of rc, sequential threshold
// binning (exactly mirrors the reference scan, integer-exact since
// t = k*C/NS = 64k), then out[b, k*C + c] = F[b,c] / (hk[k+1]-hk[k]).
// Unset edges stay 0 -> division yields inf, matching reference semantics.
// ---------------------------------------------------------------------------
__global__ void __launch_bounds__(256) finalize_kernel(const float* __restrict__ F,  // [B*C]
                                                       const int*   __restrict__ rc, // [B*H]
                                                       float*       __restrict__ out) {
    __shared__ int   sHc[H_];
    __shared__ float shkd[NS];
    const int b   = blockIdx.x;
    const int tid = threadIdx.x;

    if (tid == 0) {
        // exclusive cumsum: Hc[0]=0, Hc[j] = sum rc[b,0..j-1]
        sHc[0] = 0;
        int acc = 0;
        for (int j = 1; j < H_; ++j) { acc += rc[b * H_ + j - 1]; sHc[j] = acc; }

        float hk[NS + 1];
        for (int k = 0; k < NS; ++k) hk[k] = 0.0f;
        hk[NS] = (float)H_;
        int k = 1;
        for (int j = 1; j <= H_ - 2; ++j) {            // j in [1, H-2]
            int t = (k * C_) / NS;                     // = 64*k, exact
            if (k < NS && sHc[j] <= t && sHc[j + 1] > t) {
                hk[k] = (float)j;
                ++k;
            }
        }
        for (int kk = 0; kk < NS; ++kk) shkd[kk] = hk[kk + 1] - hk[kk];
    }
    __syncthreads();

    // out[b, k*C + c] = F[b,c] / width[k];  NS*C = 4096 per b
    for (int i = tid; i < NS * C_; i += 256) {
        int k = i >> 9;          // /C_
        int c = i & (C_ - 1);
        out[(size_t)b * (NS * C_) + i] = F[b * C_ + c] / shkd[k];
    }
}

// ---------------------------------------------------------------------------
extern "C" void kernel_launch(void* const* d_in, const int* in_sizes, int n_in,
                              void* d_out, int out_size, void* d_ws, size_t ws_size,
                              hipStream_t stream) {
    const float* x   = (const float*)d_in[0];      // [16,512,128,128] f32
    float*       out = (float*)d_out;              // [16, 4096] f32

    // workspace layout: F [B*C] floats, then rc [B*H] ints
    float* Fws = (float*)d_ws;
    int*   rc  = (int*)((char*)d_ws + (size_t)B_ * C_ * sizeof(float));

    zero_rc_kernel<<<(B_ * H_ + 255) / 256, 256, 0, stream>>>(rc, B_ * H_);

    size_t shmem = HW_ * sizeof(float) + H_ * sizeof(int) + 256 * sizeof(float);
    stats_kernel<<<B_ * C_, 256, shmem, stream>>>(x, Fws, rc);

    finalize_kernel<<<B_, 256, 0, stream>>>(Fws, rc, out);
}